// TangentManifoldVAE_84061099918077
// MI455X (gfx1250) — compile-verified
//
#include <hip/hip_runtime.h>
#include <hip/hip_bf16.h>

typedef float v8f __attribute__((ext_vector_type(8)));
typedef float v2f __attribute__((ext_vector_type(2)));

#define B_   32
#define D_   2048
#define H_   512
#define L_   32
#define N_   256
#define K_   5
#define TLEN 1024   // B_*L_

// ---------------------------------------------------------------------------
// fp32 WMMA GEMM, 32x32 register tile per wave (2x2 of 16x16 accumulators).
// C[m,n] = sum_k A(m,k)*B(k,n) (+bias[n]) (optional clip to ±1)
// A(m,k) = A[m*As_r + k*As_c]; B(k,n) = B[k*Bs_r + n*Bs_c]  -> free transposes.
// ACONT: As_c==1 (A fragment = one b64 load); BCONT: Bs_r==1 (same for B).
// K stepped by 8 = two v_wmma_f32_16x16x4_f32 rounds; 8 WMMAs per loop body.
// ---------------------------------------------------------------------------
template<bool ACONT, bool BCONT>
__global__ void wmma_gemm32(const float* __restrict__ A, int As_r, int As_c,
                            const float* __restrict__ Bm, int Bs_r, int Bs_c,
                            float* __restrict__ C, int ldc,
                            const float* __restrict__ bias,
                            int M, int N, int K, int clip)
{
    const int wavesPerBlock = blockDim.x >> 5;
    int wave   = blockIdx.x * wavesPerBlock + (threadIdx.x >> 5);
    int tilesN = N >> 5, tilesM = M >> 5;
    if (wave >= tilesM * tilesN) return;          // wave-uniform guard (EXEC stays all-1)
    int tm   = wave / tilesN, tn = wave - tm * tilesN;
    int lane = threadIdx.x & 31;
    int half = lane >> 4;                         // selects K pair {0,1}/{2,3} and row halves
    int m16  = lane & 15;

    const float* Arow0 = A  + (size_t)((tm << 5) + m16) * As_r;
    const float* Arow1 = Arow0 + (size_t)16 * As_r;
    const float* Bcol0 = Bm + (size_t)((tn << 5) + m16) * Bs_c;
    const float* Bcol1 = Bcol0 + (size_t)16 * Bs_c;

    v8f acc00 = {0.f,0.f,0.f,0.f,0.f,0.f,0.f,0.f};
    v8f acc01 = acc00, acc10 = acc00, acc11 = acc00;

    for (int k0 = 0; k0 < K; k0 += 8) {
#pragma unroll
        for (int kk = 0; kk < 2; ++kk) {
            int ka = k0 + (kk << 2) + (half << 1);
            v2f a0, a1, b0, b1;
            if (ACONT) {
                a0 = *(const v2f*)(Arow0 + ka);
                a1 = *(const v2f*)(Arow1 + ka);
            } else {
                a0.x = Arow0[(size_t)ka * As_c];  a0.y = Arow0[(size_t)(ka + 1) * As_c];
                a1.x = Arow1[(size_t)ka * As_c];  a1.y = Arow1[(size_t)(ka + 1) * As_c];
            }
            if (BCONT) {
                b0 = *(const v2f*)(Bcol0 + ka);
                b1 = *(const v2f*)(Bcol1 + ka);
            } else {
                b0.x = Bcol0[(size_t)ka * Bs_r];  b0.y = Bcol0[(size_t)(ka + 1) * Bs_r];
                b1.x = Bcol1[(size_t)ka * Bs_r];  b1.y = Bcol1[(size_t)(ka + 1) * Bs_r];
            }
            acc00 = __builtin_amdgcn_wmma_f32_16x16x4_f32(false, a0, false, b0, (short)0, acc00, false, false);
            acc01 = __builtin_amdgcn_wmma_f32_16x16x4_f32(false, a0, false, b1, (short)0, acc01, false, false);
            acc10 = __builtin_amdgcn_wmma_f32_16x16x4_f32(false, a1, false, b0, (short)0, acc10, false, false);
            acc11 = __builtin_amdgcn_wmma_f32_16x16x4_f32(false, a1, false, b1, (short)0, acc11, false, false);
        }
        if (ACONT && k0 + 8 < K) __builtin_prefetch(Arow0 + k0 + 8, 0, 1);   // global_prefetch_b8
        if (BCONT && k0 + 8 < K) __builtin_prefetch(Bcol0 + k0 + 8, 0, 1);
    }

#pragma unroll
    for (int i = 0; i < 2; ++i) {
#pragma unroll
        for (int j = 0; j < 2; ++j) {
            const v8f& acc = i ? (j ? acc11 : acc10) : (j ? acc01 : acc00);
            int n = (tn << 5) + (j << 4) + m16;
            float bv = bias ? bias[n] : 0.f;
#pragma unroll
            for (int r = 0; r < 8; ++r) {
                int m = (tm << 5) + (i << 4) + r + (half << 3);  // VGPR r -> M=r / M=r+8
                float v = acc[r] + bv;
                if (clip) v = fminf(fmaxf(v, -1.f), 1.f);
                C[(size_t)m * ldc + n] = v;
            }
        }
    }
}

// ---------------------------------------------------------------------------
// BatchNorm(train, biased var) + ReLU over [32,F]; optional side outputs for
// the analytic Jacobian: normalized n, relu mask r, r*n, g*rsqrt, rho, sigma.
// One thread per feature; serial over the 32 rows -> deterministic.
// ---------------------------------------------------------------------------
__global__ void bn_relu_kernel(const float* __restrict__ a, const float* __restrict__ g,
                               const float* __restrict__ beta, int F,
                               float* __restrict__ h,  float* __restrict__ nout,
                               float* __restrict__ rout, float* __restrict__ rnout,
                               float* __restrict__ grs, float* __restrict__ rho,
                               float* __restrict__ sigma)
{
    int j = blockIdx.x * blockDim.x + threadIdx.x;
    if (j >= F) return;
    float m = 0.f;
    for (int p = 0; p < B_; ++p) m += a[p * F + j];
    m *= (1.f / B_);
    float v = 0.f;
    for (int p = 0; p < B_; ++p) { float d = a[p * F + j] - m; v += d * d; }
    v *= (1.f / B_);
    float rs = rsqrtf(v + 1e-5f);
    float gj = g[j], bj = beta[j];
    float rhoj = 0.f, sigj = 0.f;
    for (int p = 0; p < B_; ++p) {
        float nn = (a[p * F + j] - m) * rs;
        float y  = gj * nn + bj;
        float r  = (y > 0.f) ? 1.f : 0.f;
        h[p * F + j] = r * y;
        if (nout)  nout[p * F + j]  = nn;
        if (rout)  rout[p * F + j]  = r;
        if (rnout) rnout[p * F + j] = r * nn;
        rhoj += r; sigj += r * nn;
    }
    if (grs)   grs[j]   = gj * rs;
    if (rho)   rho[j]   = rhoj;
    if (sigma) sigma[j] = sigj;
}

// z = mu + eps*max(exp(0.5*clip(lvr)),1e-3); c_{b,l} gate; per-element KL part
__global__ void reparam_kernel(const float* __restrict__ mu, const float* __restrict__ lvr,
                               const float* __restrict__ eps,
                               float* __restrict__ z, float* __restrict__ cbl,
                               float* __restrict__ klpart)
{
    int t = blockIdx.x * blockDim.x + threadIdx.x;
    if (t >= TLEN) return;
    float m  = mu[t], lr = lvr[t];
    float lv = fminf(fmaxf(lr, -4.f), 4.f);
    float st = fmaxf(expf(0.5f * lv), 1e-3f);     // exp(0.5*lv) >= e^-2 so max never binds
    z[t] = m + eps[t] * st;
    float gate = (lr > -4.f && lr < 4.f) ? 1.f : 0.f;
    cbl[t] = eps[t] * 0.5f * st * gate;
    klpart[t] = -0.5f * (1.f + lv - m * m - expf(lv));
}

// rows t=(b,l): w = g2rs2 * r2[b] * (Wmu[:,l] + c*Wvar[:,l]); w' = w*n2[b]; w'' = w*n2[b]^2
__global__ void build_wmat(const float* __restrict__ g2rs2, const float* __restrict__ r2,
                           const float* __restrict__ n2, const float* __restrict__ Wmu,
                           const float* __restrict__ Wvar, const float* __restrict__ cbl,
                           float* __restrict__ Wm, float* __restrict__ Wp, float* __restrict__ Wpp)
{
    int idx = blockIdx.x * blockDim.x + threadIdx.x;
    if (idx >= TLEN * H_) return;
    int t = idx >> 9, j = idx & (H_ - 1);
    int b = t >> 5, l = t & (L_ - 1);
    float w  = g2rs2[j] * r2[b * H_ + j] * (Wmu[j * L_ + l] + cbl[t] * Wvar[j * L_ + l]);
    float nb = n2[b * H_ + j];
    Wm[idx]  = w;
    Wp[idx]  = w * nb;
    Wpp[idx] = w * nb * nb;
}

// C <- C .* W2 , Chat <- Chat .* W2  (in place)
__global__ void cw_mul(float* __restrict__ Cm, float* __restrict__ Ch,
                       const float* __restrict__ W2)
{
    int i = blockIdx.x * blockDim.x + threadIdx.x;
    if (i >= H_ * H_) return;
    float w = W2[i];
    Cm[i] *= w;
    Ch[i] *= w;
}

// BN1-backward applied to the single nonzero batch row b, closed form.
__global__ void combine_ga1(const float* __restrict__ Am, const float* __restrict__ Pm,
                            const float* __restrict__ Qm, const float* __restrict__ Rm,
                            const float* __restrict__ r1, const float* __restrict__ n1,
                            const float* __restrict__ rho, const float* __restrict__ sigma,
                            const float* __restrict__ g1rs1, float* __restrict__ Ga1)
{
    int idx = blockIdx.x * blockDim.x + threadIdx.x;
    if (idx >= TLEN * H_) return;
    int t = idx >> 9, i = idx & (H_ - 1);
    int b = t >> 5;
    const float invB = 1.f / B_;
    float A = Am[idx], P = Pm[idx], Q = Qm[idx], R = Rm[idx];
    float r = r1[b * H_ + i], nn = n1[b * H_ + i];
    float Gbn1 = r * (A * (1.f - invB) - R * invB);
    float s  = r * A - rho[i]   * A * invB - P * invB;
    float tt = r * nn * A - sigma[i] * A * invB - Q * invB;
    Ga1[idx] = g1rs1[i] * (Gbn1 - s * invB - nn * tt * invB);
}

// ---------------- deterministic block tree-reduction helper ----------------
__device__ __forceinline__ float block_reduce_sum(float v, float* red)
{
    int tid = threadIdx.x;
    red[tid] = v;
    __syncthreads();
    for (int s = blockDim.x >> 1; s > 0; s >>= 1) {
        if (tid < s) red[tid] += red[tid + s];
        __syncthreads();
    }
    float r = red[0];
    __syncthreads();
    return r;
}

// |x_b|^2 (blocks 0..31) and |neighbor_n|^2 (blocks 32..287)
__global__ void row_norms(const float* __restrict__ x, const float* __restrict__ nb,
                          float* __restrict__ xnorm, float* __restrict__ nbnorm)
{
    __shared__ float red[256];
    int r = blockIdx.x;
    const float* p = (r < B_) ? (x + (size_t)r * D_) : (nb + (size_t)(r - B_) * D_);
    float acc = 0.f;
    for (int d = threadIdx.x; d < D_; d += 256) { float v = p[d]; acc += v * v; }
    float s = block_reduce_sum(acc, red);
    if (threadIdx.x == 0) { if (r < B_) xnorm[r] = s; else nbnorm[r - B_] = s; }
}

// top-5 nearest neighbors (smallest squared distance, first index on ties)
__global__ void knn_topk(const float* __restrict__ xnbT, const float* __restrict__ xnorm,
                         const float* __restrict__ nbnorm, int* __restrict__ idx)
{
    __shared__ float d2[N_];
    int b = blockIdx.x, tid = threadIdx.x;
    d2[tid] = xnorm[b] + nbnorm[tid] - 2.f * xnbT[b * N_ + tid];
    __syncthreads();
    if (tid == 0) {
        for (int k = 0; k < K_; ++k) {
            float best = 3.4e38f; int bi = 0;
            for (int n = 0; n < N_; ++n)
                if (d2[n] < best) { best = d2[n]; bi = n; }
            idx[b * K_ + k] = bi;
            d2[bi] = 3.4e38f;
        }
    }
}

// vec = nb[idx]-x+1e-6 ; MGS-QR of [D,5] ; T = Q^T + 1e-6   (one block per b)
__global__ void qr_tangent(const float* __restrict__ x, const float* __restrict__ nb,
                           const int* __restrict__ idx,
                           float* __restrict__ V, float* __restrict__ T)
{
    __shared__ float red[256];
    __shared__ float scal;
    int b = blockIdx.x, tid = threadIdx.x;
    for (int k = 0; k < K_; ++k) {
        int id = idx[b * K_ + k];
        float* vk = V + (size_t)(b * K_ + k) * D_;
        for (int d = tid; d < D_; d += 256)
            vk[d] = nb[(size_t)id * D_ + d] - x[(size_t)b * D_ + d] + 1e-6f;
    }
    __syncthreads();
    for (int k = 0; k < K_; ++k) {
        float* vk = V + (size_t)(b * K_ + k) * D_;
        for (int j = 0; j < k; ++j) {
            const float* qj = T + (size_t)(b * K_ + j) * D_;
            float acc = 0.f;
            for (int d = tid; d < D_; d += 256) acc += qj[d] * vk[d];
            float dot = block_reduce_sum(acc, red);
            for (int d = tid; d < D_; d += 256) vk[d] -= dot * qj[d];
            __syncthreads();
        }
        float acc = 0.f;
        for (int d = tid; d < D_; d += 256) acc += vk[d] * vk[d];
        float nrm2 = block_reduce_sum(acc, red);
        if (tid == 0) scal = rsqrtf(nrm2);
        __syncthreads();
        float inv = scal;
        float* qk = T + (size_t)(b * K_ + k) * D_;
        for (int d = tid; d < D_; d += 256) qk[d] = vk[d] * inv;
        __syncthreads();
    }
    for (int k = 0; k < K_; ++k)
        for (int d = tid; d < D_; d += 256)
            T[(size_t)(b * K_ + k) * D_ + d] += 1e-6f;
}

// per-(b,l): coef_k = g.T_k ; regpart = sum_d (g - sum_k coef_k T_k)^2
__global__ void proj_reg(const float* __restrict__ grads, const float* __restrict__ T,
                         float* __restrict__ regpart)
{
    __shared__ float red[256];
    __shared__ float coef[K_];
    int t = blockIdx.x, b = t >> 5, tid = threadIdx.x;
    const float* g = grads + (size_t)t * D_;
    for (int k = 0; k < K_; ++k) {
        const float* Tk = T + (size_t)(b * K_ + k) * D_;
        float acc = 0.f;
        for (int d = tid; d < D_; d += 256) acc += g[d] * Tk[d];
        float dot = block_reduce_sum(acc, red);
        if (tid == 0) coef[k] = dot;
    }
    __syncthreads();
    float acc = 0.f;
    for (int d = tid; d < D_; d += 256) {
        float p = 0.f;
        for (int k = 0; k < K_; ++k) p += coef[k] * T[(size_t)(b * K_ + k) * D_ + d];
        float df = g[d] - p;
        acc += df * df;
    }
    float s = block_reduce_sum(acc, red);
    if (tid == 0) regpart[t] = s;
}

// logits = h_c1 @ Wc2 + bc2  (tiny: 32x4)
__global__ void cls2_kernel(const float* __restrict__ h, const float* __restrict__ W,
                            const float* __restrict__ bias, float* __restrict__ out)
{
    int i = threadIdx.x;
    if (i >= B_ * 4) return;
    int b = i >> 2, c = i & 3;
    float s = bias[c];
    for (int j = 0; j < 256; ++j) s += h[b * 256 + j] * W[j * 4 + c];
    out[i] = s;
}

__global__ void recon_partial(const float* __restrict__ xr, const float* __restrict__ x,
                              float* __restrict__ part)
{
    __shared__ float red[256];
    int i = blockIdx.x * 256 + threadIdx.x;
    float d = xr[i] - x[i];
    float s = block_reduce_sum(d * d, red);
    if (threadIdx.x == 0) part[blockIdx.x] = s;
}

__global__ void finalize_losses(const float* __restrict__ klpart, const float* __restrict__ regpart,
                                const float* __restrict__ reconpart, float* __restrict__ out)
{
    float s = 0.f;
    for (int i = 0; i < 256; ++i) s += reconpart[i];
    out[4 * B_ + B_ * D_ + 0] = s / (float)(B_ * D_);
    float k = 0.f;
    for (int i = 0; i < TLEN; ++i) k += klpart[i];
    out[4 * B_ + B_ * D_ + 1] = fminf(k / (float)TLEN, 10.f);
    float r = 0.f;
    for (int i = 0; i < TLEN; ++i) r += regpart[i];
    out[4 * B_ + B_ * D_ + 2] = fminf(r / (float)(TLEN * D_), 10.f);
}

// ---------------------------------------------------------------------------
extern "C" void kernel_launch(void* const* d_in, const int* in_sizes, int n_in,
                              void* d_out, int out_size, void* d_ws, size_t ws_size,
                              hipStream_t stream)
{
    (void)in_sizes; (void)n_in; (void)out_size; (void)ws_size;
    const float* x    = (const float*)d_in[0];
    const float* nb   = (const float*)d_in[1];
    const float* eps  = (const float*)d_in[2];
    const float* W1   = (const float*)d_in[3];   const float* b1   = (const float*)d_in[4];
    const float* g_e1 = (const float*)d_in[5];   const float* be_e1= (const float*)d_in[6];
    const float* W2   = (const float*)d_in[7];   const float* b2   = (const float*)d_in[8];
    const float* g_e2 = (const float*)d_in[9];   const float* be_e2= (const float*)d_in[10];
    const float* Wmu  = (const float*)d_in[11];  const float* bmu  = (const float*)d_in[12];
    const float* Wvar = (const float*)d_in[13];  const float* bvar = (const float*)d_in[14];
    const float* Wd1  = (const float*)d_in[15];  const float* bd1  = (const float*)d_in[16];
    const float* g_d1 = (const float*)d_in[17];  const float* be_d1= (const float*)d_in[18];
    const float* Wd2  = (const float*)d_in[19];  const float* bd2  = (const float*)d_in[20];
    const float* g_d2 = (const float*)d_in[21];  const float* be_d2= (const float*)d_in[22];
    const float* Wd3  = (const float*)d_in[23];  const float* bd3  = (const float*)d_in[24];
    const float* Wc1  = (const float*)d_in[25];  const float* bc1  = (const float*)d_in[26];
    const float* g_c1 = (const float*)d_in[27];  const float* be_c1= (const float*)d_in[28];
    const float* Wc2  = (const float*)d_in[29];  const float* bc2  = (const float*)d_in[30];
    float* out = (float*)d_out;
    float* xrec = out + 4 * B_;                  // x_recon written directly into d_out

    float* ws = (float*)d_ws;
    size_t off = 0;
    auto alloc = [&](size_t n) -> float* { float* p = ws + off; off += n; return p; };

    float* a1   = alloc(B_ * H_);   float* n1   = alloc(B_ * H_);
    float* h1   = alloc(B_ * H_);   float* r1   = alloc(B_ * H_);
    float* rn1  = alloc(B_ * H_);
    float* g1rs1= alloc(H_); float* rho = alloc(H_); float* sigma = alloc(H_);
    float* a2   = alloc(B_ * H_);   float* n2   = alloc(B_ * H_);
    float* h2   = alloc(B_ * H_);   float* r2   = alloc(B_ * H_);
    float* g2rs2= alloc(H_);
    float* mu_  = alloc(TLEN); float* lvr_ = alloc(TLEN);
    float* z_   = alloc(TLEN); float* cbl  = alloc(TLEN); float* klpart = alloc(TLEN);
    float* ad1  = alloc(B_ * H_);   float* hd1  = alloc(B_ * H_);
    float* ad2  = alloc(B_ * H_);   float* hd2  = alloc(B_ * H_);
    float* ac1  = alloc(B_ * 256);  float* hc1  = alloc(B_ * 256);
    float* Cmat = alloc(H_ * H_);   float* Chat = alloc(H_ * H_);
    float* Wm   = alloc(TLEN * H_); float* Wp   = alloc(TLEN * H_); float* Wpp = alloc(TLEN * H_);
    float* Am   = alloc(TLEN * H_); float* Pm   = alloc(TLEN * H_);
    float* Qm   = alloc(TLEN * H_); float* Rm   = alloc(TLEN * H_);
    float* Ga1  = alloc(TLEN * H_);
    float* grads= alloc((size_t)TLEN * D_);
    float* xnbT = alloc(B_ * N_); float* xnorm = alloc(B_); float* nbnorm = alloc(N_);
    int*   idxb = (int*)alloc(B_ * K_);
    float* V    = alloc((size_t)B_ * K_ * D_);
    float* Tbuf = alloc((size_t)B_ * K_ * D_);
    float* regpart = alloc(TLEN); float* reconpart = alloc(256);

    auto gemm = [&](const float* A, int Ar, int Ac, const float* Bp, int Br, int Bc,
                    float* C, int ldc, const float* bias, int M, int N, int K, int clip) {
        int tiles  = (M / 32) * (N / 32);
        int blocks = (tiles + 3) / 4;
        if (Ac == 1 && Br == 1)
            wmma_gemm32<true, true ><<<blocks, 128, 0, stream>>>(A, Ar, Ac, Bp, Br, Bc, C, ldc, bias, M, N, K, clip);
        else if (Ac == 1)
            wmma_gemm32<true, false><<<blocks, 128, 0, stream>>>(A, Ar, Ac, Bp, Br, Bc, C, ldc, bias, M, N, K, clip);
        else if (Br == 1)
            wmma_gemm32<false, true ><<<blocks, 128, 0, stream>>>(A, Ar, Ac, Bp, Br, Bc, C, ldc, bias, M, N, K, clip);
        else
            wmma_gemm32<false, false><<<blocks, 128, 0, stream>>>(A, Ar, Ac, Bp, Br, Bc, C, ldc, bias, M, N, K, clip);
    };

    // ---------------- forward ----------------
    gemm(x, D_, 1, W1, H_, 1, a1, H_, b1, B_, H_, D_, 0);                       // a1 = x@W1+b1
    bn_relu_kernel<<<2, 256, 0, stream>>>(a1, g_e1, be_e1, H_, h1, n1, r1, rn1, g1rs1, rho, sigma);
    gemm(h1, H_, 1, W2, H_, 1, a2, H_, b2, B_, H_, H_, 0);                      // a2 = h1@W2+b2
    bn_relu_kernel<<<2, 256, 0, stream>>>(a2, g_e2, be_e2, H_, h2, n2, r2, nullptr, g2rs2, nullptr, nullptr);
    gemm(h2, H_, 1, Wmu,  L_, 1, mu_,  L_, bmu,  B_, L_, H_, 0);                // mu
    gemm(h2, H_, 1, Wvar, L_, 1, lvr_, L_, bvar, B_, L_, H_, 0);                // log-var (pre-clip)
    reparam_kernel<<<4, 256, 0, stream>>>(mu_, lvr_, eps, z_, cbl, klpart);

    gemm(z_, L_, 1, Wd1, H_, 1, ad1, H_, bd1, B_, H_, L_, 0);                   // decoder
    bn_relu_kernel<<<2, 256, 0, stream>>>(ad1, g_d1, be_d1, H_, hd1, nullptr, nullptr, nullptr, nullptr, nullptr, nullptr);
    gemm(hd1, H_, 1, Wd2, H_, 1, ad2, H_, bd2, B_, H_, H_, 0);
    bn_relu_kernel<<<2, 256, 0, stream>>>(ad2, g_d2, be_d2, H_, hd2, nullptr, nullptr, nullptr, nullptr, nullptr, nullptr);
    gemm(hd2, H_, 1, Wd3, D_, 1, xrec, D_, bd3, B_, D_, H_, 0);                 // x_recon -> d_out

    gemm(z_, L_, 1, Wc1, 256, 1, ac1, 256, bc1, B_, 256, L_, 0);                // classifier
    bn_relu_kernel<<<1, 256, 0, stream>>>(ac1, g_c1, be_c1, 256, hc1, nullptr, nullptr, nullptr, nullptr, nullptr, nullptr);
    cls2_kernel<<<1, 128, 0, stream>>>(hc1, Wc2, bc2, out);                     // logits -> d_out

    // ---------------- analytic Jacobian diagonal (grads[b,l,:]) ----------------
    gemm(r1,  1, H_, n2, H_, 1, Cmat, H_, nullptr, H_, H_, B_, 0);              // C    = r1^T @ n2
    gemm(rn1, 1, H_, n2, H_, 1, Chat, H_, nullptr, H_, H_, B_, 0);              // Chat = (r1*n1)^T @ n2
    cw_mul<<<1024, 256, 0, stream>>>(Cmat, Chat, W2);                           // ⊙ W2 in place
    build_wmat<<<2048, 256, 0, stream>>>(g2rs2, r2, n2, Wmu, Wvar, cbl, Wm, Wp, Wpp);
    gemm(Wm,  H_, 1, W2,   1, H_, Am, H_, nullptr, TLEN, H_, H_, 0);            // A = Wm  @ W2^T
    gemm(Wpp, H_, 1, W2,   1, H_, Rm, H_, nullptr, TLEN, H_, H_, 0);            // R = Wpp @ W2^T
    gemm(Wp,  H_, 1, Cmat, 1, H_, Pm, H_, nullptr, TLEN, H_, H_, 0);            // P = Wp @ CW^T
    gemm(Wp,  H_, 1, Chat, 1, H_, Qm, H_, nullptr, TLEN, H_, H_, 0);            // Q = Wp @ ChatW^T
    combine_ga1<<<2048, 256, 0, stream>>>(Am, Pm, Qm, Rm, r1, n1, rho, sigma, g1rs1, Ga1);
    gemm(Ga1, H_, 1, W1, 1, H_, grads, D_, nullptr, TLEN, D_, H_, 1);           // grads = Ga1@W1^T, clip ±1

    // ---------------- kNN + tangents + losses ----------------
    gemm(x, D_, 1, nb, 1, D_, xnbT, N_, nullptr, B_, N_, D_, 0);                // x @ neighbors^T
    row_norms<<<B_ + N_, 256, 0, stream>>>(x, nb, xnorm, nbnorm);
    knn_topk<<<B_, 256, 0, stream>>>(xnbT, xnorm, nbnorm, idxb);
    qr_tangent<<<B_, 256, 0, stream>>>(x, nb, idxb, V, Tbuf);
    proj_reg<<<TLEN, 256, 0, stream>>>(grads, Tbuf, regpart);
    recon_partial<<<256, 256, 0, stream>>>(xrec, x, reconpart);
    finalize_losses<<<1, 1, 0, stream>>>(klpart, regpart, reconpart, out);
}